// MockMACEModel_2748779070051
// MI455X (gfx1250) — compile-verified
//
#include <hip/hip_runtime.h>

typedef float v2f __attribute__((ext_vector_type(2)));
typedef float v8f __attribute__((ext_vector_type(8)));

#define NUM_GRAPHS 64

// ---------------------------------------------------------------------------
// Kernel 0: zero the scatter accumulators and the per-graph totals
// ---------------------------------------------------------------------------
__global__ void mace_zero_kernel(float* __restrict__ inter_node,
                                 float* __restrict__ zbl_node,
                                 float* __restrict__ total, int n) {
  int i = blockIdx.x * blockDim.x + threadIdx.x;
  if (i < n) {
    inter_node[i] = 0.0f;
    zbl_node[i] = 0.0f;
  }
  if (i < NUM_GRAPHS) total[i] = 0.0f;
}

// ---------------------------------------------------------------------------
// Kernel 1: per-node pass.
//  Phase 1 (wave-cooperative): node_e0 = node_attrs[16x4] @ ae via
//    V_WMMA_F32_16X16X4_F32. A-tile: lanes 0-15 load K0/K1 of row M=lane,
//    lanes 16-31 load K2/K3. B: ae replicated across all 16 columns, so
//    lanes<16 carry (ae0,ae1), lanes>=16 carry (ae2,ae3). D column N=0 is in
//    lane 0 (M=0..7) and lane 16 (M=8..15).
//  Phase 2 (per-thread): species argmax, Z, Z^0.3 packed into float4.
// ---------------------------------------------------------------------------
__global__ void mace_node_kernel(const float* __restrict__ attrs,
                                 const float* __restrict__ ae,
                                 const int* __restrict__ atomic_numbers,
                                 float* __restrict__ nodeE0,      // d_out + 64
                                 float4* __restrict__ nodeinfo,   // ws
                                 int n) {
  const int tid = blockIdx.x * blockDim.x + threadIdx.x;
  const int lane = threadIdx.x & 31;
  const int wave = tid >> 5;
  const int nwaves = (gridDim.x * blockDim.x) >> 5;

  const float ae0 = ae[0], ae1 = ae[1], ae2 = ae[2], ae3 = ae[3];

  v2f b;
  b.x = (lane < 16) ? ae0 : ae2;
  b.y = (lane < 16) ? ae1 : ae3;

  const int ntiles = n >> 4;
  for (int t = wave; t < ntiles; t += nwaves) {
    const int n0 = t << 4;
    const int m = lane & 15;
    const int koff = (lane < 16) ? 0 : 2;
    const float* ap = attrs + (size_t)(n0 + m) * 4 + koff;
    v2f a;
    a.x = ap[0];
    a.y = ap[1];
    v8f c = {};
    // D = A(16x4) x B(4x16) + 0 ; every column of D equals node_e0 tile
    v8f d = __builtin_amdgcn_wmma_f32_16x16x4_f32(
        false, a, false, b, (short)0, c, false, false);
    if (lane == 0) {
      *(float4*)(nodeE0 + n0) = make_float4(d[0], d[1], d[2], d[3]);
      *(float4*)(nodeE0 + n0 + 4) = make_float4(d[4], d[5], d[6], d[7]);
    } else if (lane == 16) {
      *(float4*)(nodeE0 + n0 + 8) = make_float4(d[0], d[1], d[2], d[3]);
      *(float4*)(nodeE0 + n0 + 12) = make_float4(d[4], d[5], d[6], d[7]);
    }
  }

  // Phase 2: species / Z metadata (+ scalar node_e0 for any ragged tail)
  const int tileEnd = ntiles << 4;
  const int stride = gridDim.x * blockDim.x;
  for (int i = tid; i < n; i += stride) {
    const float* a = attrs + (size_t)i * 4;
    const float a0 = a[0], a1 = a[1], a2 = a[2], a3 = a[3];
    int sp = 0;
    float best = a0;
    if (a1 > best) { best = a1; sp = 1; }
    if (a2 > best) { best = a2; sp = 2; }
    if (a3 > best) { best = a3; sp = 3; }
    const float Z = (float)atomic_numbers[sp];
    const float Zp = __powf(Z, 0.300f);
    nodeinfo[i] = make_float4(Z, Zp, __int_as_float(sp), 0.0f);
    if (i >= tileEnd) {
      nodeE0[i] = a0 * ae0 + a1 * ae1 + a2 * ae2 + a3 * ae3;
    }
  }
}

// ---------------------------------------------------------------------------
// Kernel 2: per-edge pass (memory/atomic bound).
// ---------------------------------------------------------------------------
__global__ void mace_edge_kernel(const float* __restrict__ pos,
                                 const int* __restrict__ ei,   // [2, nE] flat
                                 const float* __restrict__ shifts,
                                 const float4* __restrict__ nodeinfo,
                                 const float* __restrict__ prm, // 4x4
                                 float* __restrict__ inter_node,
                                 float* __restrict__ zbl_node,
                                 int nE) {
  __shared__ float s_prm[16];
  if (threadIdx.x < 16) s_prm[threadIdx.x] = prm[threadIdx.x];
  __syncthreads();

  const int e = blockIdx.x * blockDim.x + threadIdx.x;
  if (e >= nE) return;

  const int s = ei[e];
  const int r = ei[nE + e];

  const int s3 = s * 3, r3 = r * 3, e3 = e * 3;
  const float dx = pos[r3 + 0] - pos[s3 + 0] + shifts[e3 + 0];
  const float dy = pos[r3 + 1] - pos[s3 + 1] + shifts[e3 + 1];
  const float dz = pos[r3 + 2] - pos[s3 + 2] + shifts[e3 + 2];

  const float len = sqrtf(dx * dx + dy * dy + dz * dz);
  const float inv = __fdividef(1.0f, len);

  // 0.5 * (-0.1) / r scattered to receiver
  unsafeAtomicAdd(&inter_node[r], -0.05f * inv);

  const float4 ns = nodeinfo[s];
  const float4 nr = nodeinfo[r];
  const int sps = __float_as_int(ns.z);
  const int spr = __float_as_int(nr.z);
  const float rmax = s_prm[sps * 4 + spr];

  if (len < rmax) {
    // r / a  with  a = 0.4543*0.529 / (Zu^0.3 + Zv^0.3)
    const float q = len * (ns.y + nr.y) * (1.0f / (0.4543f * 0.529f));
    const float phi = 0.1818f * __expf(-3.2f * q)
                    + 0.5099f * __expf(-0.9423f * q)
                    + 0.2802f * __expf(-0.4028f * q)
                    + 0.02817f * __expf(-0.2016f * q);
    const float u = len * __fdividef(1.0f, rmax);
    const float u2 = u * u;
    const float u3 = u2 * u;
    const float u6 = u3 * u3;
    const float u7 = u6 * u;
    const float u8 = u7 * u;
    const float env = 1.0f - 28.0f * u6 + 48.0f * u7 - 21.0f * u8;
    const float v = 0.5f * 14.3996f * ns.x * nr.x * inv * phi * env;
    unsafeAtomicAdd(&zbl_node[r], v);
  }
}

// ---------------------------------------------------------------------------
// Kernel 3: finalize. node_energy = e0 + inter; graph totals via LDS staging
// (batch is sorted, so ds_add_f32 conflicts are rare) then 64 global atomics.
// ---------------------------------------------------------------------------
__global__ void mace_finalize_kernel(const float* __restrict__ inter_node,
                                     const float* __restrict__ zbl_node,
                                     const int* __restrict__ batch,
                                     float* __restrict__ nodeE,  // d_out+64
                                     float* __restrict__ total,  // d_out
                                     int n) {
  __shared__ float acc[NUM_GRAPHS];
  if (threadIdx.x < NUM_GRAPHS) acc[threadIdx.x] = 0.0f;
  __syncthreads();

  const int stride = gridDim.x * blockDim.x;
  for (int i = blockIdx.x * blockDim.x + threadIdx.x; i < n; i += stride) {
    const float e0 = nodeE[i];
    const float it = inter_node[i];
    const float zb = zbl_node[i];
    nodeE[i] = e0 + it;
    atomicAdd(&acc[batch[i]], e0 + it + zb);
  }
  __syncthreads();
  if (threadIdx.x < NUM_GRAPHS) {
    unsafeAtomicAdd(&total[threadIdx.x], acc[threadIdx.x]);
  }
}

// ---------------------------------------------------------------------------
extern "C" void kernel_launch(void* const* d_in, const int* in_sizes, int n_in,
                              void* d_out, int out_size, void* d_ws,
                              size_t ws_size, hipStream_t stream) {
  const float* positions       = (const float*)d_in[0];
  const float* node_attrs      = (const float*)d_in[1];
  const int*   edge_index      = (const int*)d_in[2];
  const int*   batch           = (const int*)d_in[3];
  const float* shifts          = (const float*)d_in[4];
  const float* atomic_energies = (const float*)d_in[5];
  const float* pair_r_max      = (const float*)d_in[6];
  const int*   atomic_numbers  = (const int*)d_in[7];

  const int nN = in_sizes[0] / 3;
  const int nE = in_sizes[2] / 2;

  float* total = (float*)d_out;             // [64]
  float* nodeE = total + NUM_GRAPHS;        // [nN]

  char* ws = (char*)d_ws;
  float* inter_node = (float*)ws;                       // [nN]
  float* zbl_node   = inter_node + nN;                  // [nN]
  size_t off = ((size_t)2 * (size_t)nN * sizeof(float) + 15) & ~(size_t)15;
  float4* nodeinfo  = (float4*)(ws + off);              // [nN]

  const int B = 256;
  const int gN = (nN + B - 1) / B;
  const int gE = (nE + B - 1) / B;

  mace_zero_kernel<<<gN, B, 0, stream>>>(inter_node, zbl_node, total, nN);
  mace_node_kernel<<<gN, B, 0, stream>>>(node_attrs, atomic_energies,
                                         atomic_numbers, nodeE, nodeinfo, nN);
  mace_edge_kernel<<<gE, B, 0, stream>>>(positions, edge_index, shifts,
                                         nodeinfo, pair_r_max, inter_node,
                                         zbl_node, nE);
  mace_finalize_kernel<<<256, B, 0, stream>>>(inter_node, zbl_node, batch,
                                              nodeE, total, nN);
}